// TranslationModel_28673201668628
// MI455X (gfx1250) — compile-verified
//
#include <hip/hip_runtime.h>
#include <math.h>

// ---------------------------------------------------------------------------
// Model dims (fixed by reference): Vin=Vout=32000, H=1024, H2=512, L=2, T=512, B=1
// ---------------------------------------------------------------------------
#define TT   512
#define HH   1024
#define H2   512
#define VV   32000
#define NWG  64
#define NTHR 256

typedef float v2f __attribute__((ext_vector_type(2)));
typedef float v8f __attribute__((ext_vector_type(8)));
typedef unsigned long long ull;

__device__ __forceinline__ float sigm(float x) { return 1.0f / (1.0f + expf(-x)); }

__device__ __forceinline__ float dot_f4(const float* __restrict__ a,
                                        const float* __restrict__ b, int n4) {
  const float4* A = (const float4*)a;
  const float4* B = (const float4*)b;
  float s = 0.0f;
  for (int i = 0; i < n4; ++i) {
    float4 u = A[i], v = B[i];
    s += u.x * v.x + u.y * v.y + u.z * v.z + u.w * v.w;
  }
  return s;
}

// monotonic float->uint key (total order preserving)
__device__ __forceinline__ unsigned flipf(float f) {
  unsigned b = __float_as_uint(f);
  return (b & 0x80000000u) ? ~b : (b | 0x80000000u);
}
__device__ __forceinline__ float unflipf(unsigned b) {
  return __uint_as_float((b & 0x80000000u) ? (b & 0x7fffffffu) : ~b);
}

// ---------------------------------------------------------------------------
// init: zero barrier state, set start token
// ---------------------------------------------------------------------------
__global__ void init_kernel(int* tokp, unsigned* bar) {
  *tokp = 1;          // START_IDX
  bar[0] = 0u;        // arrive count
  bar[1] = 0u;        // generation
}

// ---------------------------------------------------------------------------
// Embedding gather: X0[t][k] = enc_emb[x[t]][k]
// ---------------------------------------------------------------------------
__global__ __launch_bounds__(256) void embed_kernel(const int* __restrict__ x,
                                                    const float* __restrict__ emb,
                                                    float* __restrict__ X0) {
  int t = blockIdx.x;
  const float4* src = (const float4*)(emb + (size_t)x[t] * HH);
  float4* dst = (float4*)(X0 + (size_t)t * HH);
  dst[threadIdx.x] = src[threadIdx.x];
}

// ---------------------------------------------------------------------------
// Encoder input-projection GEMM (fp32 WMMA):
//   G[M=512][N=2048] = X[512][K] @ W[N][K]^T + bias[N],   K=1024
// Each wave owns a 16x16 tile, block = 8 waves => 16(M) x 128(N) per block.
// Uses V_WMMA_F32_16X16X4_F32 (full fp32 precision matrix path on CDNA5).
// ---------------------------------------------------------------------------
__global__ __launch_bounds__(256) void gemm_bias_wmma(const float* __restrict__ X,
                                                      const float* __restrict__ W,
                                                      const float* __restrict__ bias,
                                                      float* __restrict__ G) {
  const int K = 1024, N = 2048;
  int wave = threadIdx.x >> 5;
  int lane = threadIdx.x & 31;
  int n0 = blockIdx.x * 128 + wave * 16;
  int m0 = blockIdx.y * 16;
  int mr = lane & 15;
  int kh = (lane >> 4) * 2;  // lanes 0-15 carry K={0,1}; lanes 16-31 K={2,3}

  const float* xp = X + (size_t)(m0 + mr) * K + kh;  // A: row M=lane&15
  const float* wp = W + (size_t)(n0 + mr) * K + kh;  // B: col N=lane&15 (W row-major [N][K])

  v8f acc = {};
  for (int k = 0; k < K; k += 4) {
    v2f a, b;
    a.x = xp[k]; a.y = xp[k + 1];
    b.x = wp[k]; b.y = wp[k + 1];
    acc = __builtin_amdgcn_wmma_f32_16x16x4_f32(false, a, false, b, (short)0, acc,
                                                false, false);
  }

  // D layout: lane&15 = N; vgpr r holds M=r (lanes<16) or M=r+8 (lanes>=16)
  int n = n0 + mr;
  float bn = bias[n];
  int mbase = m0 + (lane >> 4) * 8;
#pragma unroll
  for (int r = 0; r < 8; ++r) G[(size_t)(mbase + r) * N + n] = acc[r] + bn;
}

// ---------------------------------------------------------------------------
// Encoder recurrence: one workgroup per direction, 1024 threads.
//   gates[t] = Gpre[t] + h @ Whh^T   (2048 outputs; thread owns 2 rows)
//   LDS-resident h[512]; c in registers of threads 0..511.
// Writes y into Y[t][dir*512 + j] and final (h,c) into hh/cc.
// ---------------------------------------------------------------------------
__global__ __launch_bounds__(1024) void enc_recur(const float* __restrict__ G,
                                                  const float* __restrict__ Whh,
                                                  float* __restrict__ Y,
                                                  float* __restrict__ hh,
                                                  float* __restrict__ cc) {
  int dir = blockIdx.x;
  const float* Gd = G + (size_t)dir * TT * 2048;
  const float* Wd = Whh + (size_t)dir * 2048 * H2;

  __shared__ float sh[H2];
  __shared__ float sg[2048];

  int tid = threadIdx.x;
  if (tid < H2) sh[tid] = 0.0f;
  float c = 0.0f, hkeep = 0.0f;
  __syncthreads();

  for (int s = 0; s < TT; ++s) {
    int t = dir ? (TT - 1 - s) : s;
    const float* gt = Gd + (size_t)t * 2048;
#pragma unroll
    for (int half = 0; half < 2; ++half) {
      int o = tid + half * 1024;
      float acc = dot_f4(Wd + (size_t)o * H2, sh, H2 / 4);
      sg[o] = acc + gt[o];
    }
    __syncthreads();
    if (tid < H2) {
      float ig = sg[tid], fg = sg[H2 + tid], gg = sg[2 * H2 + tid], og = sg[3 * H2 + tid];
      c = sigm(fg) * c + sigm(ig) * tanhf(gg);
      hkeep = sigm(og) * tanhf(c);
      Y[(size_t)t * HH + dir * H2 + tid] = hkeep;
    }
    __syncthreads();
    if (tid < H2) sh[tid] = hkeep;
    __syncthreads();
  }
  if (tid < H2) {
    hh[dir * H2 + tid] = hkeep;
    cc[dir * H2 + tid] = c;
  }
}

// ---------------------------------------------------------------------------
// Grid-wide sense barrier for the persistent decoder kernel
// ---------------------------------------------------------------------------
__device__ __forceinline__ void grid_sync(unsigned* cnt, unsigned* gen) {
  __syncthreads();
  if (threadIdx.x == 0) {
    __threadfence();
    unsigned g = ((volatile unsigned*)gen)[0];
    if (atomicAdd(cnt, 1u) == NWG - 1) {
      atomicExch(cnt, 0u);
      __threadfence();
      atomicAdd(gen, 1u);
    } else {
      while (((volatile unsigned*)gen)[0] == g) { __builtin_amdgcn_s_sleep(2); }
    }
    __threadfence();
  }
  __syncthreads();
}

__device__ __forceinline__ ull block_max_u64(ull v, ull* s) {
  s[threadIdx.x] = v;
  __syncthreads();
  for (int off = NTHR / 2; off > 0; off >>= 1) {
    if (threadIdx.x < off) {
      ull o = s[threadIdx.x + off];
      if (o > s[threadIdx.x]) s[threadIdx.x] = o;
    }
    __syncthreads();
  }
  ull r = s[0];
  __syncthreads();
  return r;
}

__device__ __forceinline__ float block_sum_f32(float v, float* s) {
  s[threadIdx.x] = v;
  __syncthreads();
  for (int off = NTHR / 2; off > 0; off >>= 1) {
    if (threadIdx.x < off) s[threadIdx.x] += s[threadIdx.x + off];
    __syncthreads();
  }
  float r = s[0];
  __syncthreads();
  return r;
}

// ---------------------------------------------------------------------------
// Persistent decoder: 64 WGs x 256 threads (16384 threads), 512 greedy steps.
// Per step:
//   gates_l = x@Wih_l^T + h_l@Whh_l^T + b_l  (4096 dots, 4 threads/dot)
//   cell update (wg0); logits = h1@clf_W^T + clf_b (2 dots/thread);
//   softmax (deterministic per-block partials) + argmax (u64 keys) -> next tok
// ---------------------------------------------------------------------------
__global__ __launch_bounds__(NTHR) void decoder_persistent(
    const float* __restrict__ dec_emb, const float* __restrict__ dec_Wih,
    const float* __restrict__ dec_Whh, const float* __restrict__ dec_b,
    const float* __restrict__ clf_W, const float* __restrict__ clf_b,
    float* __restrict__ h, float* __restrict__ c, float* __restrict__ gates,
    float* __restrict__ logits, ull* __restrict__ maxPart,
    float* __restrict__ sumPart, int* __restrict__ tokp, unsigned* bar_cnt,
    unsigned* bar_gen, float* __restrict__ out) {
  __shared__ ull s64[NTHR];
  __shared__ float s32[NTHR];

  int gtid = blockIdx.x * NTHR + threadIdx.x;  // 0..16383
  int o4 = gtid >> 2;                          // gate row 0..4095
  int sub = gtid & 3;                          // k-chunk 0..3 (256 each)

  for (int t = 0; t < TT; ++t) {
    int tok = ((volatile int*)tokp)[0];

    // ---- two LSTM layers ----
    for (int l = 0; l < 2; ++l) {
      const float* x = (l == 0) ? (dec_emb + (size_t)tok * HH) : h;
      const float* hl = h + l * HH;
      const float* wi = dec_Wih + (size_t)l * 4096 * HH + (size_t)o4 * HH + sub * 256;
      const float* wh = dec_Whh + (size_t)l * 4096 * HH + (size_t)o4 * HH + sub * 256;
      float p = dot_f4(wi, x + sub * 256, 64) + dot_f4(wh, hl + sub * 256, 64);
      p += __shfl_xor(p, 1);
      p += __shfl_xor(p, 2);
      if (sub == 0) gates[l * 4096 + o4] = p + dec_b[l * 4096 + o4];
      grid_sync(bar_cnt, bar_gen);

      if (blockIdx.x == 0) {
        for (int j = threadIdx.x; j < HH; j += NTHR) {
          float ig = gates[l * 4096 + j];
          float fg = gates[l * 4096 + 1024 + j];
          float gg = gates[l * 4096 + 2048 + j];
          float og = gates[l * 4096 + 3072 + j];
          float cj = c[l * HH + j];
          cj = sigm(fg) * cj + sigm(ig) * tanhf(gg);
          c[l * HH + j] = cj;
          h[l * HH + j] = sigm(og) * tanhf(cj);
        }
      }
      grid_sync(bar_cnt, bar_gen);
    }

    // ---- logits (2 rows per thread) + argmax/max key ----
    const float* h1 = h + HH;
    int o1 = gtid;
    int o2 = gtid + 16384;
    if (o2 < VV) __builtin_prefetch(clf_W + (size_t)o2 * HH, 0, 1);
    float l1 = clf_b[o1] + dot_f4(clf_W + (size_t)o1 * HH, h1, 256);
    logits[o1] = l1;
    ull best = ((ull)flipf(l1) << 32) | (ull)(0xFFFFFFFFu - (unsigned)o1);
    if (o2 < VV) {
      float l2 = clf_b[o2] + dot_f4(clf_W + (size_t)o2 * HH, h1, 256);
      logits[o2] = l2;
      ull k2 = ((ull)flipf(l2) << 32) | (ull)(0xFFFFFFFFu - (unsigned)o2);
      if (k2 > best) best = k2;
    }
    best = block_max_u64(best, s64);
    if (threadIdx.x == 0) maxPart[blockIdx.x] = best;
    grid_sync(bar_cnt, bar_gen);

    // ---- exp pass ----
    ull gkey = 0;
#pragma unroll 4
    for (int i = 0; i < NWG; ++i) {
      ull v = maxPart[i];
      if (v > gkey) gkey = v;
    }
    float maxv = unflipf((unsigned)(gkey >> 32));
    float e1 = expf(logits[o1] - maxv);
    logits[o1] = e1;
    float local = e1;
    if (o2 < VV) {
      float e2 = expf(logits[o2] - maxv);
      logits[o2] = e2;
      local += e2;
    }
    float bs = block_sum_f32(local, s32);
    if (threadIdx.x == 0) sumPart[blockIdx.x] = bs;
    grid_sync(bar_cnt, bar_gen);

    // ---- normalize, emit y[:,t], pick next token ----
    float total = 0.0f;
#pragma unroll 4
    for (int i = 0; i < NWG; ++i) total += sumPart[i];
    float inv = 1.0f / total;
    out[(size_t)o1 * TT + t] = logits[o1] * inv;
    if (o2 < VV) out[(size_t)o2 * TT + t] = logits[o2] * inv;
    if (blockIdx.x == 0 && threadIdx.x == 0)
      *tokp = (int)(0xFFFFFFFFu - (unsigned)(gkey & 0xFFFFFFFFull));
    grid_sync(bar_cnt, bar_gen);
  }
}

// ---------------------------------------------------------------------------
// Host-side orchestration
// ---------------------------------------------------------------------------
extern "C" void kernel_launch(void* const* d_in, const int* in_sizes, int n_in,
                              void* d_out, int out_size, void* d_ws, size_t ws_size,
                              hipStream_t stream) {
  (void)in_sizes; (void)n_in; (void)out_size; (void)ws_size;
  const int*   x       = (const int*)d_in[0];
  const float* enc_emb = (const float*)d_in[1];
  const float* enc_Wih = (const float*)d_in[2];
  const float* enc_Whh = (const float*)d_in[3];
  const float* enc_b   = (const float*)d_in[4];
  const float* dec_emb = (const float*)d_in[5];
  const float* dec_Wih = (const float*)d_in[6];
  const float* dec_Whh = (const float*)d_in[7];
  const float* dec_b   = (const float*)d_in[8];
  const float* clf_W   = (const float*)d_in[9];
  const float* clf_b   = (const float*)d_in[10];
  float* out = (float*)d_out;

  // workspace layout (float offsets; all 16B aligned)
  float* w = (float*)d_ws;
  float* X0      = w;                 // [512*1024]
  float* X1      = w + 524288;        // [512*1024]
  float* G       = w + 1048576;       // [2][512*2048]
  float* hh      = w + 3145728;       // [2*1024]
  float* cc      = w + 3147776;       // [2*1024]
  float* gates   = w + 3149824;       // [2*4096]
  float* logits  = w + 3158016;       // [32000]
  float* sumPart = w + 3190016;       // [64]
  ull*   maxPart = (ull*)(w + 3190080);   // [64] (8B aligned)
  int*   tokp    = (int*)(w + 3190208);
  unsigned* bar  = (unsigned*)(w + 3190209);  // cnt, gen

  init_kernel<<<1, 1, 0, stream>>>(tokp, bar);
  embed_kernel<<<TT, 256, 0, stream>>>(x, enc_emb, X0);

  for (int l = 0; l < 2; ++l) {
    const float* Xin = l ? X1 : X0;
    float* Y = l ? X0 : X1;  // layer-1 outputs are dead; reuse X0 as scratch
    for (int d = 0; d < 2; ++d) {
      gemm_bias_wmma<<<dim3(2048 / 128, TT / 16), 256, 0, stream>>>(
          Xin, enc_Wih + (size_t)(l * 2 + d) * 2048 * HH,
          enc_b + (size_t)(l * 2 + d) * 2048, G + (size_t)d * TT * 2048);
    }
    enc_recur<<<2, 1024, 0, stream>>>(G, enc_Whh + (size_t)l * 2 * 2048 * H2, Y,
                                      hh + l * HH, cc + l * HH);
  }

  decoder_persistent<<<NWG, NTHR, 0, stream>>>(
      dec_emb, dec_Wih, dec_Whh, dec_b, clf_W, clf_b, hh, cc, gates, logits,
      maxPart, sumPart, tokp, bar, bar + 1, out);
}